// GATLayer_44779329028364
// MI455X (gfx1250) — compile-verified
//
#include <hip/hip_runtime.h>
#include <cstddef>
#include <cstdint>

#define IN_DIM 128
#define OUT_DIM 128
#define NEG_SLOPE 0.2f

typedef float v2f __attribute__((ext_vector_type(2)));
typedef float v8f __attribute__((ext_vector_type(8)));
typedef __attribute__((address_space(3))) float lds_f32;

__device__ __forceinline__ float leaky(float x) {
    return x > 0.0f ? x : NEG_SLOPE * x;
}

// ---------------------------------------------------------------------------
// Kernel 0: fold attention vector through W2:
//   vvec[0:128]   = a_src @ W2   (so s_src = h @ vvec_src)
//   vvec[128:256] = a_dst @ W2
// ---------------------------------------------------------------------------
__global__ void prep_attn_kernel(const float* __restrict__ W2,
                                 const float* __restrict__ attn,
                                 float* __restrict__ vvec) {
    int j = threadIdx.x;  // 0..127
    float vs = 0.0f, vd = 0.0f;
    for (int k = 0; k < OUT_DIM; ++k) {
        float w = W2[k * IN_DIM + j];
        vs += attn[k] * w;
        vd += attn[OUT_DIM + k] * w;
    }
    vvec[j] = vs;
    vvec[IN_DIM + j] = vd;
}

// ---------------------------------------------------------------------------
// Kernel 1: init accumulators. out := 0, segmax := -inf bits, segsum := 0.
// ---------------------------------------------------------------------------
__global__ void init_kernel(float* __restrict__ out,
                            int* __restrict__ segmax,
                            float* __restrict__ segsum,
                            int n_nodes, size_t total) {
    size_t i = (size_t)blockIdx.x * blockDim.x + threadIdx.x;
    if (i < total) out[i] = 0.0f;
    if (i < (size_t)n_nodes) {
        segmax[i] = (int)0xFF800000;  // -inf as float bits
        segsum[i] = 0.0f;
    }
}

// ---------------------------------------------------------------------------
// Kernel 2: h_trans = h @ W1^T  via V_WMMA_F32_16X16X4_F32 (full fp32).
// Block = 256 threads = 8 waves; block owns 16 rows, wave w owns cols
// [16w, 16w+16). The block's A tile (16x128 f32 = 8KB, contiguous in memory)
// is staged into LDS ONCE via gfx1250 async-to-LDS (ASYNCcnt path) instead of
// being loaded 8x redundantly by every wave; WMMA A operands then come from
// LDS (ds_load_b64). B (W1 rows) loads stay global (L2-resident, 64KB).
// N_NODES = 50000 = 3125 * 16 exactly -> no tail, EXEC all ones for WMMA.
// ---------------------------------------------------------------------------
__global__ __launch_bounds__(256) void gemm_wmma_kernel(
        const float* __restrict__ h, const float* __restrict__ W1,
        float* __restrict__ h_trans) {
    __shared__ float a_tile[16 * IN_DIM];  // 8 KB

    const int t    = threadIdx.x;          // 0..255
    const int wave = t >> 5;
    const int lane = t & 31;
    const int l15  = lane & 15;
    const int kh   = (lane >> 4) * 2;      // 0 or 2
    const int m0   = blockIdx.x * 16;
    const int n0   = wave * 16;

    // --- async stage A tile: 2048 floats = 512 x b128; 2 transfers/thread ---
    {
        const unsigned lds_base = (unsigned)(unsigned long long)(lds_f32*)a_tile;
        const float* gsrc = h + (size_t)m0 * IN_DIM + t * 4;   // 16B aligned
        unsigned loff = lds_base + (unsigned)t * 16u;
        asm volatile("global_load_async_to_lds_b128 %0, %1, off"
                     :: "v"(loff), "v"(gsrc) : "memory");
        const float* gsrc2 = gsrc + 16 * IN_DIM / 2;           // second 4KB half
        asm volatile("global_load_async_to_lds_b128 %0, %1, off"
                     :: "v"(loff + 4096u), "v"(gsrc2) : "memory");
        asm volatile("s_wait_asynccnt 0x0" ::: "memory");
    }
    __syncthreads();

    const float* __restrict__ brow = W1 + (size_t)(n0 + l15) * IN_DIM + kh;
    const float* arow = &a_tile[l15 * IN_DIM + kh];

    v8f acc = {};
#pragma unroll 8
    for (int k = 0; k < IN_DIM; k += 4) {
        v2f a = *(const v2f*)(arow + k);   // ds_load_b64
        v2f b = *(const v2f*)(brow + k);   // global_load_b64
        acc = __builtin_amdgcn_wmma_f32_16x16x4_f32(
            /*neg_a=*/false, a, /*neg_b=*/false, b,
            /*c_mod=*/(short)0, acc, /*reuse_a=*/false, /*reuse_b=*/false);
    }

    const int mrow = m0 + (lane >> 4) * 8;
#pragma unroll
    for (int r = 0; r < 8; ++r) {
        h_trans[(size_t)(mrow + r) * OUT_DIM + n0 + l15] = acc[r];
    }
}

// ---------------------------------------------------------------------------
// Kernel 3: per-node scores. One wave per node; lane handles 4 channels.
//   s_src[i] = h[i,:] . vvec[0:128] ; s_dst[i] = h[i,:] . vvec[128:256]
// ---------------------------------------------------------------------------
__global__ __launch_bounds__(256) void score_kernel(
        const float* __restrict__ h, const float* __restrict__ vvec,
        float* __restrict__ s_src, float* __restrict__ s_dst, int n_nodes) {
    const int wave = threadIdx.x >> 5;
    const int lane = threadIdx.x & 31;
    const int node = blockIdx.x * 8 + wave;
    if (node >= n_nodes) return;

    float4 hv = ((const float4*)(h + (size_t)node * IN_DIM))[lane];
    float4 a  = ((const float4*)vvec)[lane];
    float4 b  = ((const float4*)(vvec + IN_DIM))[lane];
    float ss = hv.x * a.x + hv.y * a.y + hv.z * a.z + hv.w * a.w;
    float sd = hv.x * b.x + hv.y * b.y + hv.z * b.z + hv.w * b.w;
#pragma unroll
    for (int off = 16; off > 0; off >>= 1) {
        ss += __shfl_down(ss, off, 32);
        sd += __shfl_down(sd, off, 32);
    }
    if (lane == 0) {
        s_src[node] = ss;
        s_dst[node] = sd;
    }
}

// ---------------------------------------------------------------------------
// Kernel 4: edge pass 1. e = leaky(s_src[src] + s_dst[dst]); seg-max over dst
// using int/uint atomic bit trick (native integer atomics, no CAS loop).
// ---------------------------------------------------------------------------
__global__ void edge_max_kernel(const int* __restrict__ src,
                                const int* __restrict__ dst,
                                const float* __restrict__ s_src,
                                const float* __restrict__ s_dst,
                                float* __restrict__ e_arr,
                                int* __restrict__ segmax, int n_edges) {
    int i = blockIdx.x * blockDim.x + threadIdx.x;
    if (i >= n_edges) return;
    float e = leaky(s_src[src[i]] + s_dst[dst[i]]);
    e_arr[i] = e;
    int* addr = segmax + dst[i];
    if (e >= 0.0f)
        atomicMax(addr, __float_as_int(e));
    else
        atomicMin((unsigned int*)addr, __float_as_uint(e));
}

// ---------------------------------------------------------------------------
// Kernel 5: edge pass 2. ex = exp(e - segmax[dst]); seg-sum over dst.
// ---------------------------------------------------------------------------
__global__ void edge_exp_kernel(const int* __restrict__ dst,
                                const float* __restrict__ e_arr,
                                const int* __restrict__ segmax,
                                float* __restrict__ ex_arr,
                                float* __restrict__ segsum, int n_edges) {
    int i = blockIdx.x * blockDim.x + threadIdx.x;
    if (i >= n_edges) return;
    int d = dst[i];
    float ex = __expf(e_arr[i] - __int_as_float(segmax[d]));
    ex_arr[i] = ex;
    atomicAdd(segsum + d, ex);
}

// ---------------------------------------------------------------------------
// Kernel 6: scatter. One wave per edge; lane handles 4 channels (float4
// gather of h_trans[src], 4 global_atomic_add_f32 into out[dst]).
// Destination (25.6 MB) is L2-resident on 192 MB L2.
// ---------------------------------------------------------------------------
__global__ __launch_bounds__(256) void edge_scatter_kernel(
        const int* __restrict__ src, const int* __restrict__ dst,
        const float* __restrict__ h_trans, const float* __restrict__ ex_arr,
        const float* __restrict__ segsum, float* __restrict__ out,
        int n_edges) {
    const int wave = threadIdx.x >> 5;
    const int lane = threadIdx.x & 31;
    const int i = blockIdx.x * 8 + wave;
    if (i >= n_edges) return;
    const int s = src[i];
    const int d = dst[i];
    const float coef = ex_arr[i] / segsum[d];
    float4 hv = ((const float4*)(h_trans + (size_t)s * OUT_DIM))[lane];
    float* o = out + (size_t)d * OUT_DIM + lane * 4;
    atomicAdd(o + 0, hv.x * coef);
    atomicAdd(o + 1, hv.y * coef);
    atomicAdd(o + 2, hv.z * coef);
    atomicAdd(o + 3, hv.w * coef);
}

// ---------------------------------------------------------------------------
// Kernel 7: finalize. out = leaky(out + bias)
// ---------------------------------------------------------------------------
__global__ void finalize_kernel(float* __restrict__ out,
                                const float* __restrict__ bias, size_t total) {
    size_t i = (size_t)blockIdx.x * blockDim.x + threadIdx.x;
    if (i >= total) return;
    out[i] = leaky(out[i] + bias[i & (OUT_DIM - 1)]);
}

// ---------------------------------------------------------------------------
extern "C" void kernel_launch(void* const* d_in, const int* in_sizes, int n_in,
                              void* d_out, int out_size, void* d_ws, size_t ws_size,
                              hipStream_t stream) {
    const float* h    = (const float*)d_in[0];
    const int*   src  = (const int*)d_in[1];
    const int*   dst  = (const int*)d_in[2];
    const float* W1   = (const float*)d_in[3];
    const float* W2   = (const float*)d_in[4];
    // d_in[5] = W3 : dead code in the reference (result never used)
    const float* attn = (const float*)d_in[6];
    const float* bias = (const float*)d_in[7];
    float* out = (float*)d_out;

    const int n_nodes = in_sizes[0] / IN_DIM;   // 50000
    const int n_edges = in_sizes[1];            // 800000
    const size_t total = (size_t)n_nodes * OUT_DIM;

    // Workspace layout (all fp32): ~33 MB
    float* ws       = (float*)d_ws;
    float* h_trans  = ws;                                   // n_nodes*128
    float* s_src    = h_trans + total;                      // n_nodes
    float* s_dst    = s_src + n_nodes;                      // n_nodes
    int*   segmax   = (int*)(s_dst + n_nodes);              // n_nodes
    float* segsum   = (float*)(segmax + n_nodes);           // n_nodes
    float* e_arr    = segsum + n_nodes;                     // n_edges
    float* ex_arr   = e_arr + n_edges;                      // n_edges
    float* vvec     = ex_arr + n_edges;                     // 256

    // 0) fold attn through W2
    prep_attn_kernel<<<1, 128, 0, stream>>>(W2, attn, vvec);

    // 1) init accumulators (also zeroes d_out -> deterministic per call)
    {
        int blocks = (int)((total + 255) / 256);
        init_kernel<<<blocks, 256, 0, stream>>>(out, segmax, segsum, n_nodes, total);
    }

    // 2) h_trans = h @ W1^T  (fp32 WMMA, async-LDS staged A)
    gemm_wmma_kernel<<<n_nodes / 16, 256, 0, stream>>>(h, W1, h_trans);

    // 3) node scores
    score_kernel<<<(n_nodes + 7) / 8, 256, 0, stream>>>(h, vvec, s_src, s_dst, n_nodes);

    // 4) edge leaky + segment max
    edge_max_kernel<<<(n_edges + 255) / 256, 256, 0, stream>>>(
        src, dst, s_src, s_dst, e_arr, segmax, n_edges);

    // 5) edge exp + segment sum
    edge_exp_kernel<<<(n_edges + 255) / 256, 256, 0, stream>>>(
        dst, e_arr, segmax, ex_arr, segsum, n_edges);

    // 6) weighted scatter-sum (the bandwidth-dominant phase)
    edge_scatter_kernel<<<(n_edges + 7) / 8, 256, 0, stream>>>(
        src, dst, h_trans, ex_arr, segsum, out, n_edges);

    // 7) bias + leaky
    finalize_kernel<<<(int)((total + 255) / 256), 256, 0, stream>>>(out, bias, total);
}